// PAVAE_73237782332001
// MI455X (gfx1250) — compile-verified
//
#include <hip/hip_runtime.h>
#include <hip/hip_bf16.h>

#define B_ 256
#define G_ 20000
#define P_ 1024
#define M_ 256
#define H_ 128
#define E_ 64

typedef __attribute__((ext_vector_type(2))) float v2f;
typedef __attribute__((ext_vector_type(4))) float v4f;
typedef __attribute__((ext_vector_type(8))) float v8f;

// ---------------------------------------------------------------------------
// Kernel 1: transpose x (B,G) -> xT (G,B) so the pathway gather is coalesced.
// ---------------------------------------------------------------------------
__global__ void transpose_x(const float* __restrict__ x, float* __restrict__ xT) {
    __shared__ float tile[32][33];
    int g0 = blockIdx.x * 32, b0 = blockIdx.y * 32;
    int tx = threadIdx.x, ty = threadIdx.y;
#pragma unroll
    for (int i = ty; i < 32; i += 8)
        tile[i][tx] = x[(size_t)(b0 + i) * G_ + (g0 + tx)];      // coalesced over g
    __syncthreads();
#pragma unroll
    for (int i = ty; i < 32; i += 8)
        xT[(size_t)(g0 + i) * B_ + (b0 + tx)] = tile[tx][i];     // coalesced over b
}

// ---------------------------------------------------------------------------
// Kernel 2a (fast path): a[b,p] = tanh( sum_m xT[idx[p,m], b] * w[p,m] + pb[p] )
// One block per pathway, 64 threads, each thread owns 4 batch elements via a
// single float4 load per m (512B per wave per iteration -> full-line streams).
// ---------------------------------------------------------------------------
__global__ void pathway_kernel_t(const float* __restrict__ xT,
                                 const int* __restrict__ idx,
                                 const float* __restrict__ pw,
                                 const float* __restrict__ pmask,
                                 const float* __restrict__ pb,
                                 float* __restrict__ a_out) {
    __shared__ int   s_idx[M_];
    __shared__ float s_w[M_];
    int p = blockIdx.x;
    int t = threadIdx.x;                       // 0..63; handles b = 4t..4t+3
    for (int m = t; m < M_; m += 64) {
        s_idx[m] = idx[p * M_ + m];
        s_w[m]   = pw[p * M_ + m] * pmask[p * M_ + m];
    }
    __syncthreads();
    float a0 = 0.f, a1 = 0.f, a2 = 0.f, a3 = 0.f;
#pragma unroll 4
    for (int m = 0; m < M_; ++m) {
        v4f xv = *(const v4f*)(xT + (size_t)s_idx[m] * B_ + 4 * t);  // 16B aligned
        float w = s_w[m];
        a0 = fmaf(xv.x, w, a0);
        a1 = fmaf(xv.y, w, a1);
        a2 = fmaf(xv.z, w, a2);
        a3 = fmaf(xv.w, w, a3);
    }
    float bb = pb[p];
    a_out[(size_t)(4 * t + 0) * P_ + p] = tanhf(a0 + bb);
    a_out[(size_t)(4 * t + 1) * P_ + p] = tanhf(a1 + bb);
    a_out[(size_t)(4 * t + 2) * P_ + p] = tanhf(a2 + bb);
    a_out[(size_t)(4 * t + 3) * P_ + p] = tanhf(a3 + bb);
}

// ---------------------------------------------------------------------------
// Kernel 2b (fallback, ws too small for xT): same math, strided access on x.
// ---------------------------------------------------------------------------
__global__ void pathway_kernel_s(const float* __restrict__ x,
                                 const int* __restrict__ idx,
                                 const float* __restrict__ pw,
                                 const float* __restrict__ pmask,
                                 const float* __restrict__ pb,
                                 float* __restrict__ a_out) {
    __shared__ int   s_idx[M_];
    __shared__ float s_w[M_];
    int p = blockIdx.x;
    int t = threadIdx.x;                       // 0..255 == batch b
    s_idx[t] = idx[p * M_ + t];
    s_w[t]   = pw[p * M_ + t] * pmask[p * M_ + t];
    __syncthreads();
    float acc = 0.f;
#pragma unroll 4
    for (int m = 0; m < M_; ++m)
        acc = fmaf(x[(size_t)t * G_ + s_idx[m]], s_w[m], acc);
    a_out[(size_t)t * P_ + p] = tanhf(acc + pb[p]);
}

// ---------------------------------------------------------------------------
// Kernel 3: generic f32 WMMA GEMM  D = act(A @ W + bias)  (small GEMMs)
// One wave per 16x16 D tile, V_WMMA_F32_16X16X4_F32 over the K loop.
// ---------------------------------------------------------------------------
template <int ACT>  // 0 = none, 1 = relu
__global__ void gemm_wmma_f32(const float* __restrict__ A,
                              const float* __restrict__ W,
                              const float* __restrict__ bias,
                              float* __restrict__ D,
                              int Mr, int K, int N) {
    int lane  = threadIdx.x & 31;
    int wave  = threadIdx.x >> 5;
    int waves = blockDim.x >> 5;
    int tiles_n = N >> 4;
    int tile  = blockIdx.x * waves + wave;
    int total = (Mr >> 4) * tiles_n;
    if (tile >= total) return;                 // wave-uniform exit, EXEC stays full
    int tm = tile / tiles_n;
    int tn = tile - tm * tiles_n;
    int mn   = lane & 15;
    int koff = (lane >> 4) << 1;               // 0 or 2
    const float* arow = A + (size_t)(tm * 16 + mn) * K + koff;
    const float* wcol = W + (tn * 16 + mn);
    v8f c = {};
    for (int k = 0; k < K; k += 4) {
        v2f a = *(const v2f*)(arow + k);
        v2f b;
        b.x = wcol[(size_t)(k + koff) * N];
        b.y = wcol[(size_t)(k + koff + 1) * N];
        c = __builtin_amdgcn_wmma_f32_16x16x4_f32(
                false, a, false, b, (short)0, c, false, false);
    }
    float bv = bias[tn * 16 + mn];
    int rbase = tm * 16 + ((lane >> 4) << 3);
#pragma unroll
    for (int j = 0; j < 8; ++j) {
        float v = c[j] + bv;
        if (ACT == 1) v = fmaxf(v, 0.f);
        D[(size_t)(rbase + j) * N + (tn * 16 + mn)] = v;
    }
}

// ---------------------------------------------------------------------------
// Kernel 4: decoder GEMM  recon = hd @ dec_w2 + b2   (256,128)@(128,20000)
// One 512-thread block (16 waves) per 16-column W panel: the 8KB panel is
// staged in LDS exactly once, then each wave owns one of the 16 row-tiles and
// feeds its 32-deep WMMA chain from LDS (conflict-free ds_load_b32).
// W (10MB) is read from HBM/L2 exactly once; A (128KB) streams from L2.
// ---------------------------------------------------------------------------
__global__ void dec_gemm_wmma(const float* __restrict__ A,
                              const float* __restrict__ W,
                              const float* __restrict__ bias,
                              float* __restrict__ D) {
    __shared__ float wpanel[H_ * 16];          // 128 x 16 = 8KB
    int tn = blockIdx.x;                       // 0..1249
    int t  = threadIdx.x;                      // 0..511
#pragma unroll
    for (int i = 0; i < H_ * 16; i += 512) {
        int e = i + t;
        wpanel[e] = W[(size_t)(e >> 4) * G_ + tn * 16 + (e & 15)];
    }
    __syncthreads();
    int lane = t & 31;
    int tm   = t >> 5;                         // 0..15: row tile
    int mn   = lane & 15;
    int koff = (lane >> 4) << 1;
    const float* arow = A + (size_t)(tm * 16 + mn) * H_ + koff;
    v8f c = {};
#pragma unroll
    for (int k = 0; k < H_; k += 4) {
        v2f a = *(const v2f*)(arow + k);
        v2f b;
        b.x = wpanel[(k + koff) * 16 + mn];
        b.y = wpanel[(k + koff + 1) * 16 + mn];
        c = __builtin_amdgcn_wmma_f32_16x16x4_f32(
                false, a, false, b, (short)0, c, false, false);
    }
    float bv = bias[tn * 16 + mn];
    int rbase = tm * 16 + ((lane >> 4) << 3);
#pragma unroll
    for (int j = 0; j < 8; ++j)
        D[(size_t)(rbase + j) * G_ + (tn * 16 + mn)] = c[j] + bv;
}

// ---------------------------------------------------------------------------
// Kernel 5: reparameterization (z = mean + eps*logvar, per the reference).
// ---------------------------------------------------------------------------
__global__ void reparam_kernel(const float* __restrict__ ml,
                               const float* __restrict__ eps,
                               float* __restrict__ z,
                               float* __restrict__ mean_o,
                               float* __restrict__ logvar_o) {
    int i = blockIdx.x * blockDim.x + threadIdx.x;
    if (i >= B_ * E_) return;
    int b = i >> 6, e = i & 63;                // E_ = 64
    float mean   = ml[b * (2 * E_) + e];
    float logvar = ml[b * (2 * E_) + E_ + e];
    z[i]        = fmaf(eps[i], logvar, mean);
    mean_o[i]   = mean;
    logvar_o[i] = logvar;
}

extern "C" void kernel_launch(void* const* d_in, const int* in_sizes, int n_in,
                              void* d_out, int out_size, void* d_ws, size_t ws_size,
                              hipStream_t stream) {
    const float* x     = (const float*)d_in[0];
    const int*   pidx  = (const int*)d_in[1];
    const float* pmask = (const float*)d_in[2];
    const float* eps   = (const float*)d_in[3];
    const float* pw    = (const float*)d_in[4];
    const float* pb    = (const float*)d_in[5];
    const float* ew1   = (const float*)d_in[6];
    const float* eb1   = (const float*)d_in[7];
    const float* ew2   = (const float*)d_in[8];
    const float* eb2   = (const float*)d_in[9];
    const float* dw1   = (const float*)d_in[10];
    const float* db1   = (const float*)d_in[11];
    const float* dw2   = (const float*)d_in[12];
    const float* db2   = (const float*)d_in[13];

    // workspace layout (floats)
    float* ws = (float*)d_ws;
    float* a  = ws;                 // 256*1024      = 262144
    float* h  = ws + 262144;        // 256*128       =  32768
    float* ml = ws + 294912;        // 256*128       =  32768
    float* hd = ws + 327680;        // 256*128       =  32768
    float* xT = ws + 360448;        // 20000*256     = 5120000
    size_t need_xt = (size_t)(360448 + (size_t)G_ * B_) * sizeof(float);

    // output layout: [ z (B*E) | recon (B*G) | mean (B*E) | logvar (B*E) ]
    float* out    = (float*)d_out;
    float* z_o    = out;
    float* recon  = out + (size_t)B_ * E_;
    float* mean_o = out + (size_t)B_ * E_ + (size_t)B_ * G_;
    float* logv_o = mean_o + (size_t)B_ * E_;

    if (ws_size >= need_xt) {
        transpose_x<<<dim3(G_ / 32, B_ / 32), dim3(32, 8), 0, stream>>>(x, xT);
        pathway_kernel_t<<<P_, 64, 0, stream>>>(xT, pidx, pw, pmask, pb, a);
    } else {
        pathway_kernel_s<<<P_, 256, 0, stream>>>(x, pidx, pw, pmask, pb, a);
    }

    // h = relu(a @ enc_w1 + enc_b1)       (256,1024)@(1024,128): 128 tiles
    gemm_wmma_f32<1><<<16, 256, 0, stream>>>(a, ew1, eb1, h, B_, P_, H_);
    // ml = h @ enc_w2 + enc_b2            (256,128)@(128,128):  128 tiles
    gemm_wmma_f32<0><<<16, 256, 0, stream>>>(h, ew2, eb2, ml, B_, H_, 2 * E_);
    // z / mean / logvar
    reparam_kernel<<<(B_ * E_ + 255) / 256, 256, 0, stream>>>(ml, eps, z_o, mean_o, logv_o);
    // hd = relu(z @ dec_w1 + dec_b1)      (256,64)@(64,128):    128 tiles
    gemm_wmma_f32<1><<<16, 256, 0, stream>>>(z_o, dw1, db1, hd, B_, E_, H_);
    // recon = hd @ dec_w2 + dec_b2        (256,128)@(128,20000): LDS-staged panels
    dec_gemm_wmma<<<G_ / 16, 512, 0, stream>>>(hd, dw2, db2, recon);
}